// SpikeTruncatedMixtureModel_13477607375125
// MI455X (gfx1250) — compile-verified
//
#include <hip/hip_runtime.h>

// ---------------------------------------------------------------------------
// Truncated E-step for a spike mixture model on MI455X (gfx1250, wave32).
//
// Math restructuring: with diagonal precision W = exp(-log_noise_var),
//   ll[n,k] = -0.5*(logdet + x'Wx) + G[n,u_k] + uc[u_k],   G = (x.*W) @ mu^T
//   uc[u]   = -0.5*mu_u'W mu_u + log_prop[u]
//   noise   = -0.5*(logdet + x'Wx) + NOISE_LOG_PROP
// The per-spike base -0.5*(logdet + x'Wx) is shared by ALL 11 logits (the
// noise Mahalanobis is exactly x'Wx), so it cancels under softmax. We only
// need G (a dense [n,256] fp32 GEMM -> v_wmma_f32_16x16x4_f32) and uc.
//
// Data movement: stats [256,129] accumulated in LDS per WG (132KB, CDNA5 has
// 320KB/WGP) and flushed once with global fp32 atomics; avoids ~676MB of
// global atomic traffic from the naive scatter.
// ---------------------------------------------------------------------------

typedef float v2f __attribute__((ext_vector_type(2)));
typedef float v8f __attribute__((ext_vector_type(8)));

#define D        128
#define DP1      129
#define NU       256            // number of units
#define KC       10             // candidates per spike
#define WAVES    8
#define TPB      (WAVES * 32)   // 256 threads
#define TILE_M   16             // spikes per wave per GEMM tile
#define ITERS    4
#define SPW      (WAVES * TILE_M * ITERS)   // 512 spikes per workgroup
#define NOISE_LP (-2.0f)

// LDS layout (float offsets)
#define S_STATS  0                                  // NU*DP1  = 33024
#define S_G      (NU * DP1)                         // 8*16*256 = 32768
#define S_IV     (S_G + WAVES * TILE_M * NU)        // 128
#define S_UC     (S_IV + D)                         // 256
#define S_R      (S_UC + NU)                        // 8*16*10 = 1280 (resp)
#define S_C      (S_R + WAVES * TILE_M * KC)        // 1280 (unit ids, int)
#define SMEM_F   (S_C + WAVES * TILE_M * KC)
#define SMEM_BYTES ((size_t)SMEM_F * 4)             // 274,944 B < 320KB

__global__ void zero_f32(float* p, int n) {
    int i = blockIdx.x * blockDim.x + threadIdx.x;
    if (i < n) p[i] = 0.0f;
}

__global__ __launch_bounds__(TPB, 1)
void estep_wmma(const float* __restrict__ feat,   // [n,128]
                const float* __restrict__ mu,     // [256,128]
                const float* __restrict__ lnv,    // [128]
                const float* __restrict__ lp,     // [256]
                const int*   __restrict__ cand,   // [n,10]
                float* __restrict__ stats_out,    // [256,129]
                int n)
{
    extern __shared__ float smf[];
    float* s_stats = smf + S_STATS;
    float* s_g     = smf + S_G;
    float* s_iv    = smf + S_IV;
    float* s_uc    = smf + S_UC;
    float* s_rf    = smf + S_R;
    int*   s_ci    = (int*)(smf + S_C);

    const int tid  = threadIdx.x;
    const int lane = tid & 31;
    const int wave = tid >> 5;

    // --- per-WG setup: precision diag, zero stats, per-unit constants -----
    for (int i = tid; i < D; i += TPB) s_iv[i] = __expf(-lnv[i]);
    for (int i = tid; i < NU * DP1; i += TPB) s_stats[i] = 0.0f;
    __syncthreads();
    for (int u = tid; u < NU; u += TPB) {
        const float* mr = mu + (size_t)u * D;
        float q = 0.0f;
        for (int dd = 0; dd < D; ++dd) q += mr[dd] * mr[dd] * s_iv[dd];
        s_uc[u] = -0.5f * q + lp[u];
    }
    __syncthreads();

    const int mrow = lane & 15;            // row (spike) / col (unit) in tile
    const int koff = (lane >> 4) << 1;     // 0 or 2 : K sub-offset per ISA layout
    const int wg0  = blockIdx.x * SPW;

    for (int it = 0; it < ITERS; ++it) {
        const int m0 = wg0 + (it * WAVES + wave) * TILE_M;
        if (m0 + TILE_M > n) continue;     // no barriers inside loop: safe

        // ------------------ GEMM: G[16 spikes, 256 units] -----------------
        // A (16x4 f32): v0 -> K=koff, v1 -> K=koff+1 for row M=lane&15.
        // B (4x16 f32): same K striping, N=lane&15 (layout mirror of A).
        v8f acc[NU / 16];
        #pragma unroll
        for (int t = 0; t < NU / 16; ++t) acc[t] = (v8f){0,0,0,0,0,0,0,0};

        const float* arow = feat + (size_t)(m0 + mrow) * D + koff;
        if (m0 + WAVES * TILE_M + TILE_M <= n)   // prefetch next tile's rows
            __builtin_prefetch(arow + (size_t)WAVES * TILE_M * D, 0, 0);

        #pragma unroll 4
        for (int kk = 0; kk < D / 4; ++kk) {
            v2f xr = *(const v2f*)(arow + kk * 4);
            v2f iv = *(const v2f*)(s_iv + kk * 4 + koff);
            v2f af; af.x = xr.x * iv.x; af.y = xr.y * iv.y;   // x .* W
            #pragma unroll
            for (int nt = 0; nt < NU / 16; ++nt) {
                const float* brow =
                    mu + (size_t)(nt * 16 + mrow) * D + kk * 4 + koff;
                v2f bf = *(const v2f*)brow;
                acc[nt] = __builtin_amdgcn_wmma_f32_16x16x4_f32(
                    false, af, false, bf, (short)0, acc[nt], false, false);
            }
        }

        // ---- stage G tile to this wave's LDS slice (C layout: M=v(+8)) ---
        float* gw = s_g + wave * TILE_M * NU;
        const int mhi = (lane >> 4) << 3;   // 0 or 8
        #pragma unroll
        for (int nt = 0; nt < NU / 16; ++nt)
            #pragma unroll
            for (int v = 0; v < 8; ++v)
                gw[(v + mhi) * NU + nt * 16 + mrow] = acc[nt][v];
        // same-wave LDS ops are in-order: later ds_loads see these stores.

        // ---- softmax over {G+uc} U {NOISE_LP}: lanes 0..15, one spike each
        if (lane < 16) {
            const int sp = m0 + lane;
            float lg[KC]; int uu[KC];
            float mx = NOISE_LP;
            #pragma unroll
            for (int k = 0; k < KC; ++k) {
                uu[k] = cand[(size_t)sp * KC + k];
                lg[k] = gw[lane * NU + uu[k]] + s_uc[uu[k]];
                mx = fmaxf(mx, lg[k]);
            }
            float se = __expf(NOISE_LP - mx);
            #pragma unroll
            for (int k = 0; k < KC; ++k) { lg[k] = __expf(lg[k] - mx); se += lg[k]; }
            const float inv = 1.0f / se;
            float* rw = s_rf + (wave * TILE_M + lane) * KC;
            int*   cw = s_ci + (wave * TILE_M + lane) * KC;
            #pragma unroll
            for (int k = 0; k < KC; ++k) {
                float r = lg[k] * inv;
                rw[k] = r; cw[k] = uu[k];
                atomicAdd(&s_stats[uu[k] * DP1 + D], r);   // soft count column
            }
        }

        // ---- scatter-add resp * x: all 32 lanes per (spike,cand) pair ----
        for (int m = 0; m < TILE_M; ++m) {
            const float* xr = feat + (size_t)(m0 + m) * D;
            float x0 = xr[lane], x1 = xr[lane + 32],
                  x2 = xr[lane + 64], x3 = xr[lane + 96];
            const float* rw = s_rf + (wave * TILE_M + m) * KC;
            const int*   cw = s_ci + (wave * TILE_M + m) * KC;
            #pragma unroll
            for (int k = 0; k < KC; ++k) {
                float r = rw[k];                 // broadcast ds_load
                float* srow = s_stats + cw[k] * DP1;
                atomicAdd(&srow[lane],      r * x0);
                atomicAdd(&srow[lane + 32], r * x1);
                atomicAdd(&srow[lane + 64], r * x2);
                atomicAdd(&srow[lane + 96], r * x3);
            }
        }
    }

    __syncthreads();
    // ---- flush per-WG partial stats with global fp32 atomics -------------
    for (int i = tid; i < NU * DP1; i += TPB)
        atomicAdd(&stats_out[i], s_stats[i]);
}

extern "C" void kernel_launch(void* const* d_in, const int* in_sizes, int n_in,
                              void* d_out, int out_size, void* d_ws, size_t ws_size,
                              hipStream_t stream) {
    const float* feat = (const float*)d_in[0];
    const float* mu   = (const float*)d_in[1];
    const float* lnv  = (const float*)d_in[2];
    const float* lp   = (const float*)d_in[3];
    const int*   cand = (const int*)d_in[4];
    float* out = (float*)d_out;

    const int n = in_sizes[0] / D;          // 131072 spikes

    zero_f32<<<(out_size + 255) / 256, 256, 0, stream>>>(out, out_size);

    const int grid = (n + SPW - 1) / SPW;   // 256 workgroups
    estep_wmma<<<grid, TPB, SMEM_BYTES, stream>>>(feat, mu, lnv, lp, cand, out, n);
}